// TransformerWithCache_55173149884971
// MI455X (gfx1250) — compile-verified
//
#include <hip/hip_runtime.h>
#include <math.h>

#define D_DIM   1024
#define S_LEN   4096
#define L_LAYERS 24
#define ABLOCKS 32
#define AWAVES  (ABLOCKS * 8)   // 256 partial softmax waves

typedef float v2f __attribute__((ext_vector_type(2)));
typedef float v8f __attribute__((ext_vector_type(8)));

__global__ void copy_vec_kernel(const float* __restrict__ src, float* __restrict__ dst, int n) {
    int i = blockIdx.x * blockDim.x + threadIdx.x;
    if (i < n) dst[i] = src[i];
}

// y = act(x @ W + bias [+ resid]) with W [1024x1024] row-major (row = input dim).
// grid 8 x 256 threads; each wave computes 16 output columns via V_WMMA_F32_16X16X4_F32:
//   A row0 = x[i0..i0+3] (rows 1..15 zero), B = 4x16 weight tile, K=1024 in 256 wmma steps.
// Software-pipelined by 8: issue 16 global B-loads + 8 LDS A-loads as clauses, then run the
// 8-deep WMMA accumulate chain, so loads overlap the matrix pipe instead of full-draining
// loadcnt/dscnt before every wmma (what round-1 codegen did with the exec-masked variant).
// A-lane masking is done with a multiply by sel (1.0 for lanes 0/16, else 0) - no exec writes,
// so EXEC stays all-ones as WMMA requires.
__global__ void matvec_wmma_kernel(const float* __restrict__ x,
                                   const float* __restrict__ W,
                                   const float* __restrict__ bias,
                                   const float* __restrict__ resid,
                                   float* __restrict__ out,
                                   int relu_flag) {
    __shared__ float xs[D_DIM];
    for (int i = threadIdx.x; i < D_DIM; i += 256) xs[i] = x[i];
    __syncthreads();

    const int lane = threadIdx.x & 31;
    const int wave = threadIdx.x >> 5;
    const int j0   = (blockIdx.x * 8 + wave) * 16;
    const int half = lane >> 4;   // 0: rows K0/K1, 1: rows K2/K3
    const int col  = lane & 15;
    const float sel = (col == 0) ? 1.f : 0.f;

    const float* wp = W + (size_t)(2 * half) * D_DIM + j0 + col;
    v8f c = {};
    for (int i0 = 0; i0 < D_DIM; i0 += 32) {   // 8 wmma steps per trip
        float bx[8], by[8];
        #pragma unroll
        for (int u = 0; u < 8; u++) {
            bx[u] = wp[0];
            by[u] = wp[D_DIM];
            wp += 4 * D_DIM;
        }
        float ax[8], ay[8];
        #pragma unroll
        for (int u = 0; u < 8; u++) {
            ax[u] = xs[i0 + 4 * u + 2 * half] * sel;
            ay[u] = xs[i0 + 4 * u + 2 * half + 1] * sel;
        }
        #pragma unroll
        for (int u = 0; u < 8; u++) {
            v2f a; a.x = ax[u]; a.y = ay[u];
            v2f b; b.x = bx[u]; b.y = by[u];
            c = __builtin_amdgcn_wmma_f32_16x16x4_f32(false, a, false, b, (short)0, c,
                                                      false, false);
        }
    }
    if (lane < 16) {
        int j = j0 + lane;
        float y = c[0] + bias[j];   // D VGPR0, lanes 0-15 = (M=0, N=0..15)
        if (resid) y += resid[j];
        if (relu_flag) y = fmaxf(y, 0.f);
        out[j] = y;
    }
}

// Flash-style attention partials. cache_l = cache + l*2*S*1024 ([0]=V, [1]=K).
// qkv_l = [q(1024) | k_i(1024) | v_i(1024)]. Wave g owns positions s = g, g+256, ...
// Lane covers head h = lane/2, dims (lane&1)*32..+31 (qbase = lane*32). K/V each read once;
// mask = any(v[s]!=0) over ALL heads comes free from __any across the 32 lanes.
__global__ void attn_partial_kernel(const float* __restrict__ cache_l,
                                    const float* __restrict__ qkv_l,
                                    float* __restrict__ pm, float* __restrict__ pl,
                                    float* __restrict__ pacc) {
    const int lane = threadIdx.x & 31;
    const int wave = threadIdx.x >> 5;
    const int g = blockIdx.x * 8 + wave;
    const int qbase = lane * 32;

    float q[32];
    #pragma unroll
    for (int j = 0; j < 32; j++) q[j] = qkv_l[qbase + j] * 0.125f;  // 1/sqrt(64) folded in

    float m = -INFINITY, lsum = 0.f;
    float acc[32];
    #pragma unroll
    for (int j = 0; j < 32; j++) acc[j] = 0.f;

    for (int s = g; s < S_LEN; s += AWAVES) {
        const float *kp, *vp;
        if (s < S_LEN - 1) {   // shifted cache: new pos s reads old pos s+1
            kp = cache_l + ((size_t)(S_LEN + s + 1)) * D_DIM + qbase;
            vp = cache_l + ((size_t)(s + 1)) * D_DIM + qbase;
        } else {               // last position = freshly computed k_i / v_i
            kp = qkv_l + D_DIM + qbase;
            vp = qkv_l + 2 * D_DIM + qbase;
        }
        float kv[32], vv[32];
        #pragma unroll
        for (int j = 0; j < 8; j++) {
            float4 k4 = ((const float4*)kp)[j];
            kv[4*j] = k4.x; kv[4*j+1] = k4.y; kv[4*j+2] = k4.z; kv[4*j+3] = k4.w;
            float4 v4 = ((const float4*)vp)[j];
            vv[4*j] = v4.x; vv[4*j+1] = v4.y; vv[4*j+2] = v4.z; vv[4*j+3] = v4.w;
        }
        int nz = 0;
        #pragma unroll
        for (int j = 0; j < 32; j++) nz |= (vv[j] != 0.f);
        float part = 0.f;
        #pragma unroll
        for (int j = 0; j < 32; j++) part += q[j] * kv[j];
        float score = part + __shfl_xor(part, 1, 32);   // pair lanes of same head
        if (__any(nz)) {                                 // position mask over all H*HD
            float mn = fmaxf(m, score);
            float scale = expf(m - mn);                  // exp(-inf - finite) = 0 on first hit
            float p = expf(score - mn);
            lsum = lsum * scale + p;
            #pragma unroll
            for (int j = 0; j < 32; j++) acc[j] = acc[j] * scale + p * vv[j];
            m = mn;
        }
    }
    int base = g * 32 + lane;
    pm[base] = m; pl[base] = lsum;
    #pragma unroll
    for (int j = 0; j < 8; j++) {
        float4 a4 = make_float4(acc[4*j], acc[4*j+1], acc[4*j+2], acc[4*j+3]);
        ((float4*)(pacc + (size_t)base * 32))[j] = a4;
    }
}

// Merge the 256 per-wave partials -> vals[1024]. Thread d owns dim d (lane=d/32, i=d%32).
__global__ void attn_combine_kernel(const float* __restrict__ pm, const float* __restrict__ pl,
                                    const float* __restrict__ pacc, float* __restrict__ vals) {
    int d = blockIdx.x * 256 + threadIdx.x;
    int lane = d >> 5, i = d & 31;
    float M = -INFINITY;
    for (int w = 0; w < AWAVES; w++) M = fmaxf(M, pm[w * 32 + lane]);
    float L = 0.f, V = 0.f;
    for (int w = 0; w < AWAVES; w++) {
        float mw = pm[w * 32 + lane];
        if (mw != -INFINITY) {
            float e = expf(mw - M);
            L += e * pl[w * 32 + lane];
            V += e * pacc[(size_t)(w * 32 + lane) * 32 + i];
        }
    }
    vals[d] = V / L;
}

// Two-pass layernorm over 1024 (matches jnp.mean / biased jnp.var, eps=1e-6).
__global__ void layernorm_kernel(const float* __restrict__ t, const float* __restrict__ sc,
                                 const float* __restrict__ bi, float* __restrict__ xout) {
    __shared__ float red[256];
    __shared__ float s_mean, s_rstd;
    float v[4];
    #pragma unroll
    for (int j = 0; j < 4; j++) v[j] = t[threadIdx.x + 256 * j];
    float s = v[0] + v[1] + v[2] + v[3];
    red[threadIdx.x] = s; __syncthreads();
    for (int k = 128; k > 0; k >>= 1) { if (threadIdx.x < k) red[threadIdx.x] += red[threadIdx.x + k]; __syncthreads(); }
    if (threadIdx.x == 0) s_mean = red[0] * (1.f / 1024.f);
    __syncthreads();
    float mean = s_mean, ss = 0.f;
    #pragma unroll
    for (int j = 0; j < 4; j++) { float dd = v[j] - mean; ss += dd * dd; }
    red[threadIdx.x] = ss; __syncthreads();
    for (int k = 128; k > 0; k >>= 1) { if (threadIdx.x < k) red[threadIdx.x] += red[threadIdx.x + k]; __syncthreads(); }
    if (threadIdx.x == 0) s_rstd = rsqrtf(red[0] * (1.f / 1024.f) + 1e-6f);
    __syncthreads();
    float rstd = s_rstd;
    #pragma unroll
    for (int j = 0; j < 4; j++) {
        int idx = threadIdx.x + 256 * j;
        xout[idx] = (v[j] - mean) * rstd * sc[idx] + bi[idx];
    }
}

// Final x copy + the three tiny logit heads.
__global__ void heads_kernel(const float* __restrict__ x,
                             const float* __restrict__ Wpi, const float* __restrict__ bpi,
                             const float* __restrict__ Wvh, const float* __restrict__ bvh,
                             const float* __restrict__ Wc,  const float* __restrict__ bc,
                             float* __restrict__ out) {
    int t = threadIdx.x;
    #pragma unroll
    for (int j = 0; j < 4; j++) out[t + 256 * j] = x[t + 256 * j];
    if (t < 32) {
        float a = bpi[t];
        for (int i = 0; i < D_DIM; i++) a += x[i] * Wpi[i * 32 + t];
        out[1024 + t] = a;
    } else if (t < 39) {
        int j = t - 32; float a = bvh[j];
        for (int i = 0; i < D_DIM; i++) a += x[i] * Wvh[i * 7 + j];
        out[1056 + j] = a;
    } else if (t < 47) {
        int j = t - 39; float a = bc[j];
        for (int i = 0; i < D_DIM; i++) a += x[i] * Wc[i * 8 + j];
        out[1063 + j] = a;
    }
}

// new_cache[l][c][s] = old_cache[l][c][s+1] for s < S-1; last row from stored v_i/k_i.
// 4 elements per thread (stride 256 within a 1024-chunk): perfectly coalesced b32 loads and
// stores with 4x the outstanding-request ILP per thread. 805 MB read + 805 MB write.
__global__ void cache_shift_kernel(const float* __restrict__ cache,
                                   const float* __restrict__ qkv,
                                   float* __restrict__ out_cache) {
    long long base = (long long)blockIdx.x * 1024 + threadIdx.x;
    #pragma unroll
    for (int u = 0; u < 4; u++) {
        long long idx = base + u * 256;
        int r = (int)(idx & 1023);
        long long row = idx >> 10;
        int s = (int)(row & (S_LEN - 1));
        int c = (int)((row >> 12) & 1);
        int l = (int)(row >> 13);
        float val;
        if (s < S_LEN - 1) {
            val = cache[((((long long)l * 2 + c) << 12) + (s + 1)) * D_DIM + r];
        } else {
            val = qkv[l * 3072 + (c ? D_DIM : 2 * D_DIM) + r];  // c==0 -> v_i, c==1 -> k_i
        }
        out_cache[idx] = val;
    }
}

extern "C" void kernel_launch(void* const* d_in, const int* in_sizes, int n_in,
                              void* d_out, int out_size, void* d_ws, size_t ws_size,
                              hipStream_t stream) {
    (void)in_sizes; (void)n_in; (void)out_size; (void)ws_size;
    const float* x_in  = (const float*)d_in[0];
    const float* cache = (const float*)d_in[1];
    const float* Wv  = (const float*)d_in[2];  const float* bv  = (const float*)d_in[3];
    const float* Wq  = (const float*)d_in[4];  const float* bq  = (const float*)d_in[5];
    const float* Wk  = (const float*)d_in[6];  const float* bk  = (const float*)d_in[7];
    const float* Wo  = (const float*)d_in[8];  const float* bo  = (const float*)d_in[9];
    const float* l1s = (const float*)d_in[10]; const float* l1b = (const float*)d_in[11];
    const float* l2s = (const float*)d_in[12]; const float* l2b = (const float*)d_in[13];
    const float* Wf1 = (const float*)d_in[14]; const float* bf1 = (const float*)d_in[15];
    const float* Wf2 = (const float*)d_in[16]; const float* bf2 = (const float*)d_in[17];
    const float* Wpi = (const float*)d_in[18]; const float* bpi = (const float*)d_in[19];
    const float* Wvh = (const float*)d_in[20]; const float* bvh = (const float*)d_in[21];
    const float* Wc  = (const float*)d_in[22]; const float* bc  = (const float*)d_in[23];
    float* out = (float*)d_out;

    float* ws   = (float*)d_ws;
    float* X    = ws;                 // 1024 current activations
    float* TMP  = ws + 1024;          // 1024 pre-LN vector
    float* HID  = ws + 2048;          // 1024 FF hidden
    float* VALS = ws + 3072;          // 1024 attention output
    float* QKV  = ws + 4096;          // 24 * 3072 (q|k|v per layer; k_i/v_i feed cache writer)
    float* PM   = QKV + L_LAYERS * 3072;   // 256*32
    float* PL   = PM + AWAVES * 32;        // 256*32
    float* PACC = PL + AWAVES * 32;        // 256*32*32

    copy_vec_kernel<<<4, 256, 0, stream>>>(x_in, X, 1024);

    for (int l = 0; l < L_LAYERS; l++) {
        size_t w_off = (size_t)l * D_DIM * D_DIM;
        size_t b_off = (size_t)l * D_DIM;
        float* q = QKV + l * 3072;
        matvec_wmma_kernel<<<8, 256, 0, stream>>>(X, Wq + w_off, bq + b_off, nullptr, q, 0);
        matvec_wmma_kernel<<<8, 256, 0, stream>>>(X, Wk + w_off, bk + b_off, nullptr, q + 1024, 0);
        matvec_wmma_kernel<<<8, 256, 0, stream>>>(X, Wv + w_off, bv + b_off, nullptr, q + 2048, 0);
        attn_partial_kernel<<<ABLOCKS, 256, 0, stream>>>(
            cache + (size_t)l * 2 * S_LEN * D_DIM, q, PM, PL, PACC);
        attn_combine_kernel<<<4, 256, 0, stream>>>(PM, PL, PACC, VALS);
        matvec_wmma_kernel<<<8, 256, 0, stream>>>(VALS, Wo + w_off, bo + b_off, X, TMP, 0);
        layernorm_kernel<<<1, 256, 0, stream>>>(TMP, l1s + b_off, l1b + b_off, X);
        matvec_wmma_kernel<<<8, 256, 0, stream>>>(X, Wf1 + w_off, bf1 + b_off, nullptr, HID, 1);
        matvec_wmma_kernel<<<8, 256, 0, stream>>>(HID, Wf2 + w_off, bf2 + b_off, X, TMP, 0);
        layernorm_kernel<<<1, 256, 0, stream>>>(TMP, l2s + b_off, l2b + b_off, X);
    }

    heads_kernel<<<1, 256, 0, stream>>>(X, Wpi, bpi, Wvh, bvh, Wc, bc, out);
    cache_shift_kernel<<<(L_LAYERS * 2 * S_LEN * D_DIM) / 1024, 256, 0, stream>>>(
        cache, QKV, out + 1071);
}